// RatiosOnlyAdaFastfood_36601711296836
// MI455X (gfx1250) — compile-verified
//
#include <hip/hip_runtime.h>
#include <math.h>

// RatiosOnlyAdaFastfood for MI455X (gfx1250, wave32).
// Memory-bound (~384MB @ 23.3 TB/s). mean_T commutes with the fixed linear
// chain -> sum deltas once, run ONE transform per row.
// FWHT-4096 = H16 (x) H16 (x) H16 -> three passes of 16x16 tile GEMMs with
// exact-precision V_WMMA_F32_16X16X4_F32. H16 operands (+/-1, Sylvester
// order) are generated once per lane, pinned in VGPRs via empty asm so the
// compiler cannot rematerialize them inside the WMMA passes.

typedef __attribute__((ext_vector_type(2))) float v2f;
typedef __attribute__((ext_vector_type(8))) float v8f;

#define FF_D 4096          // transform length (16*16*16)

// Sylvester-order Hadamard entry: H16[i][j] = (-1)^popc(i&j).
__device__ __forceinline__ float hval(int i, int j) {
    return (__popc(i & j) & 1) ? -1.0f : 1.0f;
}

// D = M (16x16 tile in LDS, row stride 16) x H16.  h = per-lane H operand.
__device__ __forceinline__ v8f tile_MxH(const float* __restrict__ tile,
                                        const v2f* __restrict__ h, int n, int g) {
    v8f acc = {0.f,0.f,0.f,0.f,0.f,0.f,0.f,0.f};
#pragma unroll
    for (int j = 0; j < 4; ++j) {
        const int k = 4 * j + 2 * g;
        v2f a;
        a.x = tile[n * 16 + k];        // A[m=n][k]
        a.y = tile[n * 16 + k + 1];
        acc = __builtin_amdgcn_wmma_f32_16x16x4_f32(
            false, a, false, h[j], (short)0, acc, false, false);
    }
    return acc;
}

// D = H16 x M, M's element [k][n] at base[k*stride + n] in LDS.
__device__ __forceinline__ v8f tile_HxM(const float* __restrict__ base, int stride,
                                        const v2f* __restrict__ h, int n, int g) {
    v8f acc = {0.f,0.f,0.f,0.f,0.f,0.f,0.f,0.f};
#pragma unroll
    for (int j = 0; j < 4; ++j) {
        const int k = 4 * j + 2 * g;
        v2f b;
        b.x = base[k * stride + n];    // B[k][n]
        b.y = base[(k + 1) * stride + n];
        acc = __builtin_amdgcn_wmma_f32_16x16x4_f32(
            false, h[j], false, b, (short)0, acc, false, false);
    }
    return acc;
}

// Unnormalized FWHT of 4096 f32 in LDS, viewed as x[a][b][c], i=a*256+b*16+c.
// Passes C and B share the per-'a' tile partition (same wave owns both), so
// only pass A needs workgroup barriers. 8 waves x 2 tiles each.
__device__ __forceinline__ void fwht4096(float* X, const v2f* h, int wave, int lane) {
    const int n = lane & 15;
    const int g = lane >> 4;
    __syncthreads();
#pragma unroll
    for (int tt = 0; tt < 2; ++tt) {
        float* tile = X + (wave + 8 * tt) * 256;
        v8f rc = tile_MxH(tile, h, n, g);            // along c: tile = tile*H
#pragma unroll
        for (int v = 0; v < 8; ++v) tile[(v + 8 * g) * 16 + n] = rc[v];
        v8f rb = tile_HxM(tile, 16, h, n, g);        // along b: tile = H*tile
#pragma unroll
        for (int v = 0; v < 8; ++v) tile[(v + 8 * g) * 16 + n] = rb[v];
    }
    __syncthreads();
#pragma unroll
    for (int tt = 0; tt < 2; ++tt) {
        float* col = X + (wave + 8 * tt) * 16;       // along a: fixed b column-tile
        v8f ra = tile_HxM(col, 256, h, n, g);
#pragma unroll
        for (int v = 0; v < 8; ++v) col[(v + 8 * g) * 256 + n] = ra[v];
    }
    __syncthreads();
}

__global__ __launch_bounds__(256)
void RatiosOnlyAdaFastfood_kernel(const float* __restrict__ pre,
                                  const float* __restrict__ deltas,
                                  const float* __restrict__ theta,
                                  const float* __restrict__ Bsign,
                                  const float* __restrict__ G,
                                  const int*   __restrict__ Pi,
                                  float* __restrict__ out,
                                  int rows, int T) {
    __shared__ __align__(16) float X[FF_D];
    __shared__ __align__(16) float Y[FF_D];
    const int r    = blockIdx.x;
    const int t    = threadIdx.x;
    const int wave = t >> 5;     // wave32
    const int lane = t & 31;
    const int n    = lane & 15;
    const int g    = lane >> 4;
    const size_t rowoff = (size_t)r * FF_D;

    // Per-lane H16 operand pairs: identical layout serves as A-of(HxM) and
    // B-of(MxH). Pin in VGPRs so clang cannot rematerialize per use.
    v2f h[4];
#pragma unroll
    for (int j = 0; j < 4; ++j) {
        const int k = 4 * j + 2 * g;
        h[j].x = hval(n, k);
        h[j].y = hval(n, k + 1);
    }
    asm volatile("" : "+v"(h[0]), "+v"(h[1]), "+v"(h[2]), "+v"(h[3]));

    // Hint the final read of pre's row into cache early (global_prefetch_b8).
    __builtin_prefetch(pre + rowoff + (size_t)t * 16, 0, 0);

    float4* __restrict__       X4 = (float4*)X;
    float4* __restrict__       Y4 = (float4*)Y;
    const float4* __restrict__ B4 = (const float4*)Bsign;
    const float4* __restrict__ G4 = (const float4*)G;
    const int4*  __restrict__  P4 = (const int4*)Pi;
    const float4* __restrict__ pre4 = (const float4*)(pre + rowoff);
    float4* __restrict__       out4 = (float4*)(out + rowoff);

    // Stage 0: X = (sum_t deltas[t][r]) * B   (1/T deferred to final scale)
    {
        const size_t rstride4 = (size_t)rows * (FF_D / 4);
        const float4* d4 = (const float4*)(deltas) + (size_t)r * (FF_D / 4);
#pragma unroll
        for (int j = 0; j < FF_D / 1024; ++j) {
            const int i = t + 256 * j;
            float4 acc = make_float4(0.f, 0.f, 0.f, 0.f);
            for (int tt = 0; tt < T; ++tt) {
                const float4 v = d4[(size_t)tt * rstride4 + i];
                acc.x += v.x; acc.y += v.y; acc.z += v.z; acc.w += v.w;
            }
            const float4 b = B4[i];
            X4[i] = make_float4(acc.x * b.x, acc.y * b.y, acc.z * b.z, acc.w * b.w);
        }
    }
    fwht4096(X, h, wave, lane);

    // Stage 1: gather-permute with G:  Y[i] = G[i] * X[Pi[i]]
#pragma unroll
    for (int j = 0; j < FF_D / 1024; ++j) {
        const int i = t + 256 * j;
        const int4   p  = P4[i];
        const float4 gg = G4[i];
        Y4[i] = make_float4(gg.x * X[p.x], gg.y * X[p.y],
                            gg.z * X[p.z], gg.w * X[p.w]);
    }
    fwht4096(Y, h, wave, lane);

    // Stage 2: soft ratio mask  s[i] = sigmoid(10*(ratio*D - i))
    {
        const float th    = theta[0];
        const float ratio = 1.0f / (1.0f + __expf(-th)) * 0.95f + 0.05f; // EPS=0.05
        const float c     = ratio * (float)FF_D;
#pragma unroll
        for (int j = 0; j < FF_D / 1024; ++j) {
            const int i  = t + 256 * j;
            const float i0 = (float)(4 * i);
            float4 v = Y4[i];
            v.x *= 1.0f / (1.0f + __expf(-10.0f * (c - i0)));
            v.y *= 1.0f / (1.0f + __expf(-10.0f * (c - (i0 + 1.0f))));
            v.z *= 1.0f / (1.0f + __expf(-10.0f * (c - (i0 + 2.0f))));
            v.w *= 1.0f / (1.0f + __expf(-10.0f * (c - (i0 + 3.0f))));
            Y4[i] = v;
        }
    }
    fwht4096(Y, h, wave, lane);

    // Stage 3: scatter-permute (== gather with inv_Pi):  X[Pi[j]] = Y[j] * G[j]
#pragma unroll
    for (int j = 0; j < FF_D / 1024; ++j) {
        const int i = t + 256 * j;
        const int4   p  = P4[i];
        const float4 gg = G4[i];
        const float4 v  = Y4[i];
        X[p.x] = v.x * gg.x;
        X[p.y] = v.y * gg.y;
        X[p.z] = v.z * gg.z;
        X[p.w] = v.w * gg.w;
    }
    fwht4096(X, h, wave, lane);

    // Final: out = pre + X * B * (64^-4 / T)   (four orthonormal FWHTs + mean)
    const float scale = 1.0f / (16777216.0f * (float)T);
#pragma unroll
    for (int j = 0; j < FF_D / 1024; ++j) {
        const int i = t + 256 * j;
        const float4 p = pre4[i];
        const float4 b = B4[i];
        const float4 x = X4[i];
        out4[i] = make_float4(p.x + x.x * b.x * scale,
                              p.y + x.y * b.y * scale,
                              p.z + x.z * b.z * scale,
                              p.w + x.w * b.w * scale);
    }
}

extern "C" void kernel_launch(void* const* d_in, const int* in_sizes, int n_in,
                              void* d_out, int out_size, void* d_ws, size_t ws_size,
                              hipStream_t stream) {
    const float* pre    = (const float*)d_in[0];
    const float* deltas = (const float*)d_in[1];
    const float* theta  = (const float*)d_in[2];
    const float* Bsign  = (const float*)d_in[3];
    const float* G      = (const float*)d_in[4];
    const int*   Pi     = (const int*)d_in[5];
    float* out = (float*)d_out;

    const int rows = in_sizes[0] / FF_D;                                // 4096
    const int T    = in_sizes[1] / (in_sizes[0] > 0 ? in_sizes[0] : 1); // 4

    RatiosOnlyAdaFastfood_kernel<<<dim3(rows), dim3(256), 0, stream>>>(
        pre, deltas, theta, Bsign, G, Pi, out, rows, T);
}